// Radon_69887707840886
// MI455X (gfx1250) — compile-verified
//
#include <hip/hip_runtime.h>
#include <math.h>

#define RES       256
#define NA        180
#define NBATCH    4
#define LSTRIDE   260          // LDS row stride in floats (256 data + 4 pad, breaks bank conflicts)
#define ANG_PER_WG 2
#define TPB       (ANG_PER_WG * RES)   // 512 threads = 16 wave32s

typedef __attribute__((ext_vector_type(4))) unsigned int uint32x4_t;
typedef __attribute__((ext_vector_type(8))) int          int32x8_t;
typedef __attribute__((ext_vector_type(4))) int          int32x4_t;

__device__ __forceinline__ float img_fetch(const float* lds, int x, int y) {
    bool ok = ((unsigned)x < (unsigned)RES) & ((unsigned)y < (unsigned)RES);
    int xc = min(max(x, 0), RES - 1);
    int yc = min(max(y, 0), RES - 1);
    float v = lds[yc * LSTRIDE + xc];
    return ok ? v : 0.0f;
}

__global__ __launch_bounds__(TPB, 1) void radon_tdm_kernel(
        const float* __restrict__ imgs,
        const float* __restrict__ angles,
        const float* __restrict__ rays,
        float* __restrict__ out) {
    extern __shared__ float lds[];
    const int tid = threadIdx.x;
    const int b   = blockIdx.y;

    // ---- TDM: DMA this batch's 256x256 fp32 image into LDS (wave 0 issues) ----
    if (tid < 32) {
        unsigned long long ga = (unsigned long long)(uintptr_t)(imgs + (size_t)b * RES * RES);
        // low 32 bits of a flat shared pointer == workgroup-relative LDS byte address
        unsigned int lds_addr = (unsigned int)(uintptr_t)(&lds[0]);

        // D# group 0: count=1 (valid), lds_addr, 57-bit global addr, type=2 ("image")
        uint32x4_t g0;
        g0.x = 1u;
        g0.y = lds_addr;
        g0.z = (unsigned int)(ga & 0xFFFFFFFFull);
        g0.w = (unsigned int)((ga >> 32) & 0x1FFFFFFull) | (2u << 30);

        // D# group 1:
        //  w0: data_size=2 (4B) | pad_enable | pad_interval=7 (256 DW) | pad_amount=3 (4 DW)
        //  dims: tensor 256x256, tile 256x256, dim0 stride 256 elements
        int32x8_t g1;
        g1[0] = (int)((2u << 16) | (1u << 20) | (7u << 22) | (3u << 25));
        g1[1] = (int)((unsigned)RES << 16);  // tensor_dim0[15:0] in bits[31:16]
        g1[2] = (int)((unsigned)RES << 16);  // tensor_dim0[31:16]=0 | tensor_dim1[15:0]
        g1[3] = (int)((unsigned)RES << 16);  // tensor_dim1[31:16]=0 | tile_dim0
        g1[4] = RES;                          // tile_dim1=256 | tile_dim2=0
        g1[5] = RES;                          // tensor_dim0_stride[31:0]
        g1[6] = 0;                            // stride0 hi | tensor_dim1_stride lo (unused, 2D)
        g1[7] = 0;

        int32x4_t g2 = {0, 0, 0, 0};          // 2D tensor: higher dims unused
        int32x4_t g3 = {0, 0, 0, 0};
        int32x8_t g4 = {0, 0, 0, 0, 0, 0, 0, 0}; // extra group (clang-23 6-arg form), unused

        __builtin_amdgcn_tensor_load_to_lds(g0, g1, g2, g3, g4, /*cpol=*/0);
        __builtin_amdgcn_s_wait_tensorcnt(0);
    }
    __syncthreads();

    // ---- per-thread ray march ----
    const int ray = tid & (RES - 1);
    const int la  = tid >> 8;                 // 0..ANG_PER_WG-1
    const int a   = blockIdx.x * ANG_PER_WG + la;

    float4 rr  = ((const float4*)rays)[ray];  // {loc, -ys, loc, +ys}
    float loc  = rr.x;
    float sy   = rr.y;
    float ey   = rr.w;
    float ang  = angles[a];
    float c, sn;
    sincosf(ang, &sn, &c);

    // rot = [[c, -sn], [sn, c]] applied to (loc, y)
    float sxr = fmaf(c,  loc, -sn * sy);
    float syr = fmaf(sn, loc,   c * sy);
    float exr = fmaf(c,  loc, -sn * ey);
    float eyr = fmaf(sn, loc,   c * ey);

    const float inv_n = 1.0f / (float)RES;
    float dx = (exr - sxr) * inv_n;
    float dy = (eyr - syr) * inv_n;
    // first sample at t = 0.5/256; image-space offset (s - 0.5) = 127.5
    float fx = fmaf(0.5f, dx, sxr) + 127.5f;
    float fy = fmaf(0.5f, dy, syr) + 127.5f;

    float acc = 0.0f;
    #pragma unroll 4
    for (int k = 0; k < RES; ++k) {
        float x0f = floorf(fx);
        float y0f = floorf(fy);
        float wx  = fx - x0f;
        float wy  = fy - y0f;
        int x0 = (int)x0f;
        int y0 = (int)y0f;

        float v00 = img_fetch(lds, x0,     y0);
        float v01 = img_fetch(lds, x0 + 1, y0);
        float v10 = img_fetch(lds, x0,     y0 + 1);
        float v11 = img_fetch(lds, x0 + 1, y0 + 1);

        float top = fmaf(wx, v01 - v00, v00);
        float bot = fmaf(wx, v11 - v10, v10);
        acc = fmaf(wy, bot - top, acc + top);

        fx += dx;
        fy += dy;
    }

    float length = ey - sy;                   // |end - start| = 2*ys
    out[((size_t)b * NA + a) * RES + ray] = acc * length * inv_n;
}

extern "C" void kernel_launch(void* const* d_in, const int* in_sizes, int n_in,
                              void* d_out, int out_size, void* d_ws, size_t ws_size,
                              hipStream_t stream) {
    const float* imgs   = (const float*)d_in[0];
    const float* angles = (const float*)d_in[1];
    const float* rays   = (const float*)d_in[2];
    float* out = (float*)d_out;

    dim3 grid(NA / ANG_PER_WG, NBATCH);       // 90 x 4 = 360 workgroups
    dim3 block(TPB);                          // 512 threads
    size_t smem = (size_t)LSTRIDE * RES * sizeof(float);  // 266,240 B < 320 KB/WGP
    radon_tdm_kernel<<<grid, block, smem, stream>>>(imgs, angles, rays, out);
}